// UniformKernel_7516192768992
// MI455X (gfx1250) — compile-verified
//
#include <hip/hip_runtime.h>

// Moving-average (K=10, replicate padding) over the bin axis + per-column
// normalization.  Pure bandwidth problem (~0.5 flop/byte): ~768MB traffic ->
// ~30us floor at 23.3 TB/s.  CDNA5 path: async global->LDS (ASYNCcnt) ring
// pipeline with b128 transfers, global_prefetch on the reduction pass.
// 4 columns per thread => ~0.02 wave-instructions per HBM byte (budget ~0.11).

#define KWIN 10
#define COLS_PER_BLOCK 256      // threads per block
#define VEC 4                   // columns per thread (float4 / b128 path)
#define ROW_CHUNKS 16
#define RING_SLOTS 16           // LDS ring rows per wave (power of two)
#define PF_DEPTH 12             // async rows in flight per wave

typedef __attribute__((ext_vector_type(4))) float fv4;

#define WAIT_ASYNC(N) asm volatile("s_wait_asynccnt %0" :: "n"(N) : "memory")

__device__ __forceinline__ void async_load_row_b128(const float* gaddr, unsigned lds_byte_off) {
  // VDST = per-lane LDS byte offset, VADDR = 64-bit per-lane global address.
  // Wave moves 32 lanes * 16B = 512B per issue.  Tracked with ASYNCcnt.
  asm volatile("global_load_async_to_lds_b128 %0, %1, off"
               :: "v"(lds_byte_off), "v"(gaddr)
               : "memory");
}

// ---------------------------------------------------------------------------
// Pass 1: partial column sums.  grid (dim/1024, ROW_CHUNKS), block 256.
// Each thread sums `rows` rows of 4 adjacent columns with b128 loads.
// ---------------------------------------------------------------------------
__global__ void colsum_partial(const float* __restrict__ x, float* __restrict__ part,
                               int nb, int dim, int rows) {
  const int c0 = (blockIdx.x * blockDim.x + threadIdx.x) * VEC;
  const int r0 = blockIdx.y * rows;
  const float* p = x + (size_t)r0 * dim + c0;

  fv4 s0 = {0.f, 0.f, 0.f, 0.f};
  fv4 s1 = {0.f, 0.f, 0.f, 0.f};
  for (int j = 0; j < rows; j += 2) {
    int jp = r0 + j + 32; if (jp >= nb) jp = nb - 1;          // clamped prefetch
    __builtin_prefetch(x + (size_t)jp * dim + c0, 0, 1);      // -> global_prefetch_b8
    s0 += *(const fv4*)(p + (size_t)(j + 0) * dim);
    s1 += *(const fv4*)(p + (size_t)(j + 1) * dim);
  }
  *(fv4*)&part[(size_t)blockIdx.y * dim + c0] = s0 + s1;
}

// ---------------------------------------------------------------------------
// Pass 2: combine partials (fixed order => deterministic), add replicate-pad
// edge corrections, produce inv[d] = 1 / (width * Wsum[d]).
// Wsum = 10*S + 5x0 -4x1 -3x2 -2x3 -1x4 -1x[nb-4] -2x[nb-3] -3x[nb-2] +11x[nb-1]
// ---------------------------------------------------------------------------
__global__ void combine_inv(const float* __restrict__ x, const float* __restrict__ part,
                            const float* __restrict__ bins, float* __restrict__ inv,
                            int nb, int dim, int nchunks) {
  const int c0 = (blockIdx.x * blockDim.x + threadIdx.x) * VEC;
  fv4 s = {0.f, 0.f, 0.f, 0.f};
  for (int c = 0; c < nchunks; ++c) s += *(const fv4*)&part[(size_t)c * dim + c0];
  fv4 wsum = 10.f * s;
  wsum += 5.f * (*(const fv4*)&x[c0])
        - 4.f * (*(const fv4*)&x[(size_t)1 * dim + c0])
        - 3.f * (*(const fv4*)&x[(size_t)2 * dim + c0])
        - 2.f * (*(const fv4*)&x[(size_t)3 * dim + c0])
        - 1.f * (*(const fv4*)&x[(size_t)4 * dim + c0]);
  const size_t base = (size_t)(nb - 4) * dim + c0;
  wsum += -1.f * (*(const fv4*)&x[base])
          -2.f * (*(const fv4*)&x[base + (size_t)dim])
          -3.f * (*(const fv4*)&x[base + (size_t)2 * dim])
         +11.f * (*(const fv4*)&x[base + (size_t)3 * dim]);
  const float width = bins[1] - bins[0];
  fv4 r;
  r.x = 1.0f / (width * wsum.x);
  r.y = 1.0f / (width * wsum.y);
  r.z = 1.0f / (width * wsum.z);
  r.w = 1.0f / (width * wsum.w);
  *(fv4*)&inv[c0] = r;
}

// ---------------------------------------------------------------------------
// Pass 3: sliding-window smoothing, async global->LDS b128 pipeline.
// grid (dim/1024, ROW_CHUNKS), block 256 = 8 waves.  Each wave owns 128
// columns (4/lane) and a private 16-row LDS ring (16 slots * 512B = 8KB/wave,
// 64KB/block of the 320KB WGP LDS).  Loads retire in issue order, so
// s_wait_asynccnt (PF-K)=2 guarantees rows <= t+10 are resident.
// W(t+1) = W(t) + x[clamp(b+6)] - x[clamp(b-4)].
// ---------------------------------------------------------------------------
__global__ void smooth_write(const float* __restrict__ x, const float* __restrict__ inv_arr,
                             float* __restrict__ out, int nb, int dim, int rows) {
  __shared__ float ring[8 * RING_SLOTS * 32 * VEC];   // 64 KB

  const int tid  = threadIdx.x;
  const int wave = tid >> 5;
  const int lane = tid & 31;
  const int c0   = (blockIdx.x * blockDim.x + tid) * VEC;
  const int r0   = blockIdx.y * rows;
  const fv4 inv  = *(const fv4*)&inv_arr[c0];

  // slot stride = 32 lanes * 4 floats = 128 floats = 512 bytes
  float* myring = &ring[wave * (RING_SLOTS * 32 * VEC) + lane * VEC];
  const unsigned lbase = (unsigned)(unsigned long long)myring;  // LDS byte offset

  const int TOT  = rows + KWIN - 1;          // row indices i = 0..TOT-1, j = r0-4+i
  const int MAIN = TOT - PF_DEPTH;           // iterations that issue a new load

  auto jclamp = [&](int i) {
    int j = r0 - 4 + i;
    j = j < 0 ? 0 : j;
    j = j >= nb ? nb - 1 : j;
    return j;
  };

  // Prime the pipeline.
#pragma unroll
  for (int i = 0; i < PF_DEPTH; ++i)
    async_load_row_b128(x + (size_t)jclamp(i) * dim + c0,
                        lbase + (unsigned)((i & (RING_SLOTS - 1)) * 32 * VEC * 4));

  WAIT_ASYNC(PF_DEPTH - KWIN);               // first 10 rows resident
  fv4 W = {0.f, 0.f, 0.f, 0.f};
#pragma unroll
  for (int i = 0; i < KWIN; ++i) W += *(const fv4*)(myring + i * 32 * VEC);

  const size_t obase = (size_t)r0 * dim + c0;
  int t = 0;
  for (; t < MAIN; ++t) {                    // steady state: store, issue, wait<=2, slide
    *(fv4*)&out[obase + (size_t)t * dim] = W * inv;
    async_load_row_b128(x + (size_t)jclamp(PF_DEPTH + t) * dim + c0,
                        lbase + (unsigned)(((PF_DEPTH + t) & (RING_SLOTS - 1)) * 32 * VEC * 4));
    WAIT_ASYNC(PF_DEPTH - KWIN);             // rows <= t+10 resident
    fv4 vin  = *(const fv4*)(myring + ((t + KWIN) & (RING_SLOTS - 1)) * 32 * VEC);
    fv4 vout = *(const fv4*)(myring + (t & (RING_SLOTS - 1)) * 32 * VEC);
    W += vin - vout;
  }
  WAIT_ASYNC(0);                             // drain; all rows resident
  for (; t < rows; ++t) {
    *(fv4*)&out[obase + (size_t)t * dim] = W * inv;
    if (t < rows - 1) {
      fv4 vin  = *(const fv4*)(myring + ((t + KWIN) & (RING_SLOTS - 1)) * 32 * VEC);
      fv4 vout = *(const fv4*)(myring + (t & (RING_SLOTS - 1)) * 32 * VEC);
      W += vin - vout;
    }
  }
}

// ---------------------------------------------------------------------------
// Fallback (tiny ws): fully fused per-column kernel, two sweeps, no scratch.
// ---------------------------------------------------------------------------
__global__ void fused_fallback(const float* __restrict__ x, const float* __restrict__ bins,
                               float* __restrict__ out, int nb, int dim) {
  const int d = blockIdx.x * blockDim.x + threadIdx.x;
  float s0 = 0.f, s1 = 0.f, s2 = 0.f, s3 = 0.f;
  for (int j = 0; j < nb; j += 4) {
    int jp = j + 32; if (jp >= nb) jp = nb - 1;
    __builtin_prefetch(x + (size_t)jp * dim + d, 0, 1);
    s0 += x[(size_t)(j + 0) * dim + d];
    s1 += x[(size_t)(j + 1) * dim + d];
    s2 += x[(size_t)(j + 2) * dim + d];
    s3 += x[(size_t)(j + 3) * dim + d];
  }
  float wsum = 10.f * ((s0 + s1) + (s2 + s3));
  wsum += 5.f * x[d] - 4.f * x[(size_t)dim + d] - 3.f * x[(size_t)2 * dim + d]
        - 2.f * x[(size_t)3 * dim + d] - x[(size_t)4 * dim + d];
  const size_t base = (size_t)(nb - 4) * dim + d;
  wsum += -x[base] - 2.f * x[base + dim] - 3.f * x[base + (size_t)2 * dim]
         + 11.f * x[base + (size_t)3 * dim];
  const float inv = 1.f / ((bins[1] - bins[0]) * wsum);

  float W = 5.f * x[d];
  for (int j = 1; j <= 5; ++j) W += x[(size_t)j * dim + d];
  for (int b = 0; b < nb; ++b) {
    out[(size_t)b * dim + d] = W * inv;
    int jin = b + 6;  jin = jin < nb ? jin : nb - 1;
    int jout = b - 4; jout = jout > 0 ? jout : 0;
    W += x[(size_t)jin * dim + d] - x[(size_t)jout * dim + d];
  }
}

extern "C" void kernel_launch(void* const* d_in, const int* in_sizes, int n_in,
                              void* d_out, int out_size, void* d_ws, size_t ws_size,
                              hipStream_t stream) {
  const float* dens = (const float*)d_in[0];
  const float* bins = (const float*)d_in[1];
  float* out = (float*)d_out;

  const int nb  = in_sizes[1];          // 2048
  const int dim = in_sizes[0] / nb;     // 32768
  const int rows = nb / ROW_CHUNKS;     // 128

  const size_t need = (size_t)(ROW_CHUNKS + 1) * (size_t)dim * sizeof(float);
  const int cpb = COLS_PER_BLOCK * VEC; // 1024 columns per block
  const bool shaped_ok = (dim % cpb) == 0 && (nb % ROW_CHUNKS) == 0 &&
                         (rows % 2) == 0 && rows >= RING_SLOTS && rows >= PF_DEPTH;

  if (ws_size >= need && shaped_ok) {
    float* part = (float*)d_ws;
    float* inv  = part + (size_t)ROW_CHUNKS * dim;
    dim3 grid2d(dim / cpb, ROW_CHUNKS);
    colsum_partial<<<grid2d, COLS_PER_BLOCK, 0, stream>>>(dens, part, nb, dim, rows);
    combine_inv<<<dim / cpb, COLS_PER_BLOCK, 0, stream>>>(
        dens, part, bins, inv, nb, dim, ROW_CHUNKS);
    smooth_write<<<grid2d, COLS_PER_BLOCK, 0, stream>>>(dens, inv, out, nb, dim, rows);
  } else {
    fused_fallback<<<dim / COLS_PER_BLOCK, COLS_PER_BLOCK, 0, stream>>>(
        dens, bins, out, nb, dim);
  }
}